// Top1Gate_61538291417526
// MI455X (gfx1250) — compile-verified
//
#include <hip/hip_runtime.h>
#include <math.h>

typedef __attribute__((ext_vector_type(2))) float v2f;
typedef __attribute__((ext_vector_type(8))) float v8f;

#define NUM_TOKENS 8192
#define MODEL_DIM  4096
#define NUM_EXPERTS 64

// ---------------------------------------------------------------------------
// K1: logits[t][e] = sum_d input[t][d] * wg[e][d]
// Wave32 WMMA f32 16x16x4. One wave owns a 16-token tile and all 4 expert
// tiles (64 experts) so each input element is loaded exactly once.
// A layout (16x4 f32): lane L -> row (L&15), K-pair = (L>>4)*2, 2 VGPRs.
// B layout (4x16 f32): lane L -> col (L&15), K-pair = (L>>4)*2, 2 VGPRs.
// C/D layout: VGPR g -> M = g + 8*(lane>=16), N = lane&15.
// ---------------------------------------------------------------------------
__global__ __launch_bounds__(128) void gemm_logits_kernel(
    const float* __restrict__ input, const float* __restrict__ wg,
    float* __restrict__ logits)
{
    const int lane  = threadIdx.x & 31;
    const int wave  = threadIdx.x >> 5;
    const int mtile = blockIdx.x * 4 + wave;     // 512 tiles of 16 tokens
    const int lidx  = lane & 15;
    const int koff  = (lane >> 4) << 1;          // 0 or 2

    const float* __restrict__ Arow = input + (size_t)(mtile * 16 + lidx) * MODEL_DIM + koff;
    const float* __restrict__ B0 = wg + (size_t)( 0 + lidx) * MODEL_DIM + koff;
    const float* __restrict__ B1 = wg + (size_t)(16 + lidx) * MODEL_DIM + koff;
    const float* __restrict__ B2 = wg + (size_t)(32 + lidx) * MODEL_DIM + koff;
    const float* __restrict__ B3 = wg + (size_t)(48 + lidx) * MODEL_DIM + koff;

    v8f c0 = {}, c1 = {}, c2 = {}, c3 = {};

    #pragma unroll 4
    for (int k = 0; k < MODEL_DIM; k += 4) {
        v2f a  = *(const v2f*)(Arow + k);
        v2f b0 = *(const v2f*)(B0 + k);
        v2f b1 = *(const v2f*)(B1 + k);
        v2f b2 = *(const v2f*)(B2 + k);
        v2f b3 = *(const v2f*)(B3 + k);
        c0 = __builtin_amdgcn_wmma_f32_16x16x4_f32(false, a, false, b0, (short)0, c0, false, false);
        c1 = __builtin_amdgcn_wmma_f32_16x16x4_f32(false, a, false, b1, (short)0, c1, false, false);
        c2 = __builtin_amdgcn_wmma_f32_16x16x4_f32(false, a, false, b2, (short)0, c2, false, false);
        c3 = __builtin_amdgcn_wmma_f32_16x16x4_f32(false, a, false, b3, (short)0, c3, false, false);
    }

    const int mbase = mtile * 16 + ((lane >> 4) ? 8 : 0);
    #pragma unroll
    for (int g = 0; g < 8; ++g) {
        float* dst = logits + (size_t)(mbase + g) * NUM_EXPERTS + lidx;
        dst[0]  = c0[g];
        dst[16] = c1[g];
        dst[32] = c2[g];
        dst[48] = c3[g];
    }
}

// ---------------------------------------------------------------------------
// K2: per-token (one wave32 per token): argmax (first-index tie-break),
// softmax denominator. gate = exp(max - max)/sum = 1/sum.
// ---------------------------------------------------------------------------
__global__ __launch_bounds__(256) void gate_rowstats_kernel(
    const float* __restrict__ logits, int* __restrict__ indices_out,
    float* __restrict__ gates_out, float* __restrict__ rowmax,
    float* __restrict__ rowsum)
{
    const int lane = threadIdx.x & 31;
    const int wave = threadIdx.x >> 5;
    const int t = blockIdx.x * 8 + wave;

    const float x0 = logits[t * NUM_EXPERTS + lane];
    const float x1 = logits[t * NUM_EXPERTS + lane + 32];

    float m; int idx;
    if (x1 > x0) { m = x1; idx = lane + 32; } else { m = x0; idx = lane; }
    #pragma unroll
    for (int off = 16; off >= 1; off >>= 1) {
        float om = __shfl_xor(m, off, 32);
        int   oi = __shfl_xor(idx, off, 32);
        if (om > m || (om == m && oi < idx)) { m = om; idx = oi; }
    }

    float s = expf(x0 - m) + expf(x1 - m);
    #pragma unroll
    for (int off = 16; off >= 1; off >>= 1) s += __shfl_xor(s, off, 32);

    if (lane == 0) {
        indices_out[t] = idx;
        gates_out[t]   = 1.0f / s;
        rowmax[t]      = m;
        rowsum[t]      = s;
    }
}

// ---------------------------------------------------------------------------
// K3: one block per expert e.
//   (a) locations1_s: in-order exclusive count of tokens choosing e
//       (ballot + popc wave scan + LDS wave-total scan), plus ce = total.
//   (b) me[e] = sum_t softmax(logits[t])[e], fixed-order block reduction.
//   Writes me_ce[e] = me[e] * ce[e]. Fully deterministic.
// ---------------------------------------------------------------------------
__global__ __launch_bounds__(256) void expert_scan_kernel(
    const float* __restrict__ logits, const int* __restrict__ indices,
    const float* __restrict__ rowmax, const float* __restrict__ rowsum,
    int* __restrict__ locations_out, float* __restrict__ me_ce)
{
    const int e    = blockIdx.x;
    const int tid  = threadIdx.x;
    const int lane = tid & 31;
    const int wave = tid >> 5;

    __shared__ unsigned wcnt[8];
    __shared__ float    red[256];

    unsigned running = 0;
    for (int base = 0; base < NUM_TOKENS; base += 256) {
        const int t = base + tid;
        const bool chosen = (indices[t] == e);
        const unsigned mask   = (unsigned)__ballot((int)chosen);  // wave32: low 32 bits
        const unsigned within = __popc(mask & ((1u << lane) - 1u));
        if (lane == 0) wcnt[wave] = __popc(mask);
        __syncthreads();
        unsigned wp = 0, tot = 0;
        #pragma unroll
        for (int w = 0; w < 8; ++w) { if (w < wave) wp += wcnt[w]; tot += wcnt[w]; }
        if (chosen) locations_out[t] = (int)(running + wp + within);
        running += tot;
        __syncthreads();
    }

    float acc = 0.0f;
    for (int t = tid; t < NUM_TOKENS; t += 256)
        acc += expf(logits[t * NUM_EXPERTS + e] - rowmax[t]) / rowsum[t];
    red[tid] = acc;
    __syncthreads();
    #pragma unroll
    for (int s = 128; s > 0; s >>= 1) {
        if (tid < s) red[tid] += red[tid + s];
        __syncthreads();
    }
    if (tid == 0) me_ce[e] = red[0] * (float)running;
}

// ---------------------------------------------------------------------------
// K4: l_aux = sum_e me_ce[e] * (E / T^2)
// ---------------------------------------------------------------------------
__global__ __launch_bounds__(64) void aux_loss_kernel(
    const float* __restrict__ me_ce, float* __restrict__ out)
{
    __shared__ float red[64];
    const int tid = threadIdx.x;
    red[tid] = me_ce[tid];
    __syncthreads();
    #pragma unroll
    for (int s = 32; s > 0; s >>= 1) {
        if (tid < s) red[tid] += red[tid + s];
        __syncthreads();
    }
    if (tid == 0)
        out[0] = red[0] * ((float)NUM_EXPERTS / ((float)NUM_TOKENS * (float)NUM_TOKENS));
}

extern "C" void kernel_launch(void* const* d_in, const int* in_sizes, int n_in,
                              void* d_out, int out_size, void* d_ws, size_t ws_size,
                              hipStream_t stream) {
    const float* input = (const float*)d_in[0];   // [8192, 4096] f32
    const float* wg    = (const float*)d_in[1];   // [64, 4096]   f32

    // d_out: [ l_aux(f32) | indices(int32 x T) | locations(int32 x T) | gates(f32 x T) ]
    float* out_f         = (float*)d_out;
    int*   indices_out   = (int*)d_out + 1;
    int*   locations_out = (int*)d_out + 1 + NUM_TOKENS;
    float* gates_out     = (float*)d_out + 1 + 2 * NUM_TOKENS;

    // workspace: logits (2 MB) + rowmax + rowsum + me_ce
    float* logits = (float*)d_ws;
    float* rowmax = logits + (size_t)NUM_TOKENS * NUM_EXPERTS;
    float* rowsum = rowmax + NUM_TOKENS;
    float* me_ce  = rowsum + NUM_TOKENS;

    gemm_logits_kernel<<<NUM_TOKENS / 64, 128, 0, stream>>>(input, wg, logits);
    gate_rowstats_kernel<<<NUM_TOKENS / 8, 256, 0, stream>>>(
        logits, indices_out, gates_out, rowmax, rowsum);
    expert_scan_kernel<<<NUM_EXPERTS, 256, 0, stream>>>(
        logits, indices_out, rowmax, rowsum, locations_out, me_ce);
    aux_loss_kernel<<<1, 64, 0, stream>>>(me_ce, out_f);
}